// MessagePassingNet_85117661872492
// MI455X (gfx1250) — compile-verified
//
#include <hip/hip_runtime.h>

typedef __attribute__((ext_vector_type(4))) int   v4i;
typedef __attribute__((ext_vector_type(4))) float v4f;
typedef __attribute__((ext_vector_type(2))) float v2f;
typedef __attribute__((ext_vector_type(8))) float v8f;

#define PENALTY_N 0.1f
#define RED1_BLOCKS 128

// ---------------------------------------------------------------------------
// Kernel 1: new_traffic starts as a copy of nodes_traffic
// ---------------------------------------------------------------------------
__global__ __launch_bounds__(256) void mp_init_kernel(
    const float* __restrict__ traffic, float* __restrict__ out, int n) {
  int i = blockIdx.x * blockDim.x + threadIdx.x;
  if (i < n) out[i] = traffic[i];
}

// ---------------------------------------------------------------------------
// Kernel 2: per-edge scatter.  transfer = |t[s]-t[d]| * n * w
//           out[s] -= transfer ; out[d] += transfer   (atomic f32, no return)
// Edge streams read with NT temporal hint (read-once, 77MB) so the hot 400KB
// traffic array stays resident in cache.  Grid-stride + prefetch-ahead.
// ---------------------------------------------------------------------------
__global__ __launch_bounds__(256) void mp_scatter_kernel(
    const int* __restrict__ src, const int* __restrict__ dst,
    const float* __restrict__ w, const float* __restrict__ traffic,
    float* __restrict__ out, int e4) {
  const v4i* __restrict__ s4p = (const v4i*)src;
  const v4i* __restrict__ d4p = (const v4i*)dst;
  const v4f* __restrict__ w4p = (const v4f*)w;

  int stride = gridDim.x * blockDim.x;
  for (int i = blockIdx.x * blockDim.x + threadIdx.x; i < e4; i += stride) {
    // prefetch next grid-stride iteration's cachelines (global_prefetch_b8)
    __builtin_prefetch((const void*)(s4p + i + stride), 0, 1);
    __builtin_prefetch((const void*)(d4p + i + stride), 0, 1);
    __builtin_prefetch((const void*)(w4p + i + stride), 0, 1);

    v4i s4 = __builtin_nontemporal_load(s4p + i);   // global_load_b128 NT
    v4i d4 = __builtin_nontemporal_load(d4p + i);
    v4f w4 = __builtin_nontemporal_load(w4p + i);

#pragma unroll
    for (int k = 0; k < 4; ++k) {
      int s = s4[k];
      int d = d4[k];
      float t = fabsf(traffic[s] - traffic[d]) * PENALTY_N * w4[k];
      __hip_atomic_fetch_add(&out[s], -t, __ATOMIC_RELAXED,
                             __HIP_MEMORY_SCOPE_AGENT);
      __hip_atomic_fetch_add(&out[d],  t, __ATOMIC_RELAXED,
                             __HIP_MEMORY_SCOPE_AGENT);
    }
  }
}

// scalar tail (only launched if E % 4 != 0; not the case for E = 6.4M)
__global__ __launch_bounds__(256) void mp_scatter_tail_kernel(
    const int* __restrict__ src, const int* __restrict__ dst,
    const float* __restrict__ w, const float* __restrict__ traffic,
    float* __restrict__ out, int start, int E) {
  int i = start + blockIdx.x * blockDim.x + threadIdx.x;
  if (i < E) {
    float t = fabsf(traffic[src[i]] - traffic[dst[i]]) * PENALTY_N * w[i];
    __hip_atomic_fetch_add(&out[src[i]], -t, __ATOMIC_RELAXED,
                           __HIP_MEMORY_SCOPE_AGENT);
    __hip_atomic_fetch_add(&out[dst[i]],  t, __ATOMIC_RELAXED,
                           __HIP_MEMORY_SCOPE_AGENT);
  }
}

// ---------------------------------------------------------------------------
// Kernel 3: per-block partial sums of  yield[i]*new_traffic[i] - cost[i]
// ---------------------------------------------------------------------------
__global__ __launch_bounds__(256) void mp_reduce1_kernel(
    const float* __restrict__ yr, const float* __restrict__ nt,
    const float* __restrict__ cost, float* __restrict__ partial, int n) {
  __shared__ float lds[256];
  int n4 = n >> 2;
  float acc = 0.0f;
  int stride = gridDim.x * blockDim.x;
  for (int i = blockIdx.x * blockDim.x + threadIdx.x; i < n4; i += stride) {
    v4f y = ((const v4f*)yr)[i];
    v4f t = ((const v4f*)nt)[i];
    v4f c = ((const v4f*)cost)[i];
    acc += (y.x * t.x - c.x) + (y.y * t.y - c.y) +
           (y.z * t.z - c.z) + (y.w * t.w - c.w);
  }
  // scalar tail (n % 4) handled by one thread
  if (blockIdx.x == 0 && threadIdx.x == 0) {
    for (int i = n4 * 4; i < n; ++i) acc += yr[i] * nt[i] - cost[i];
  }
  lds[threadIdx.x] = acc;
  __syncthreads();
  for (int s = 128; s > 0; s >>= 1) {
    if (threadIdx.x < (unsigned)s) lds[threadIdx.x] += lds[threadIdx.x + s];
    __syncthreads();
  }
  if (threadIdx.x == 0) partial[blockIdx.x] = lds[0];
}

// ---------------------------------------------------------------------------
// Kernel 4: combine <=256 partials via V_WMMA_F32_16X16X4_F32 with B = ones.
// With B all-ones, sum(D) = 16 * sum(A) + sum(C) independent of the exact
// VGPR striping of A/B, so any bijection of the padded partials into A's 64
// slots is correct.  Single block; all lanes active (EXEC==all-ones).
// ---------------------------------------------------------------------------
__global__ __launch_bounds__(256) void mp_reduce2_wmma_kernel(
    const float* __restrict__ partial, int nb, float* __restrict__ outScalar) {
  __shared__ float lds[256];
  int t = threadIdx.x;
  lds[t] = (t < nb) ? partial[t] : 0.0f;
  __syncthreads();

  int lane = t & 31;
  v2f a, b;
  a.x = lds[2 * lane]     + lds[2 * lane + 64] +
        lds[2 * lane + 128] + lds[2 * lane + 192];
  a.y = lds[2 * lane + 1]   + lds[2 * lane + 65] +
        lds[2 * lane + 129] + lds[2 * lane + 193];
  b.x = 1.0f;
  b.y = 1.0f;
  v8f c = {};
  // 8 args: (neg_a, A, neg_b, B, c_mod, C, reuse_a, reuse_b)
  v8f d = __builtin_amdgcn_wmma_f32_16x16x4_f32(
      false, a, false, b, (short)0, c, false, false);

  float s = d[0] + d[1] + d[2] + d[3] + d[4] + d[5] + d[6] + d[7];
#pragma unroll
  for (int off = 16; off > 0; off >>= 1) s += __shfl_xor(s, off, 32);

  if (t == 0) *outScalar = s * (1.0f / 16.0f);
}

// ---------------------------------------------------------------------------
extern "C" void kernel_launch(void* const* d_in, const int* in_sizes, int n_in,
                              void* d_out, int out_size, void* d_ws, size_t ws_size,
                              hipStream_t stream) {
  const int*   edge_index = (const int*)d_in[0];    // [2, E] row-major flat
  const float* edge_w     = (const float*)d_in[1];  // [E]
  const float* yieldr     = (const float*)d_in[2];  // [N]
  const float* traffic    = (const float*)d_in[3];  // [N]
  const float* cost       = (const float*)d_in[4];  // [N]
  float* out = (float*)d_out;                       // [N] new_traffic + [1] scalar

  const int E = in_sizes[1];
  const int N = in_sizes[2];
  const int* src = edge_index;
  const int* dst = edge_index + E;

  float* partial = (float*)d_ws;  // RED1_BLOCKS floats

  // 1) out[0..N) = traffic
  {
    int blocks = (N + 255) / 256;
    mp_init_kernel<<<blocks, 256, 0, stream>>>(traffic, out, N);
  }

  // 2) edge scatter (vec4 main body + scalar tail)
  {
    int e4 = E >> 2;
    int blocks = (e4 + 255) / 256;
    if (blocks > 2048) blocks = 2048;  // grid-stride, keeps prefetch useful
    if (blocks > 0)
      mp_scatter_kernel<<<blocks, 256, 0, stream>>>(src, dst, edge_w, traffic,
                                                    out, e4);
    int rem = E - e4 * 4;
    if (rem > 0)
      mp_scatter_tail_kernel<<<(rem + 255) / 256, 256, 0, stream>>>(
          src, dst, edge_w, traffic, out, e4 * 4, E);
  }

  // 3) partial reduction of yield*new_traffic - cost
  mp_reduce1_kernel<<<RED1_BLOCKS, 256, 0, stream>>>(yieldr, out, cost,
                                                     partial, N);

  // 4) WMMA-based final combine -> out[N]
  mp_reduce2_wmma_kernel<<<1, 256, 0, stream>>>(partial, RED1_BLOCKS, &out[N]);
}